// DiffusionDetMatcher_81381040324935
// MI455X (gfx1250) — compile-verified
//
#include <hip/hip_runtime.h>
#include <hip/hip_bf16.h>
#include <cfloat>
#include <cstdint>

#define N_PRED 10000
#define N_GT   1000
#define NCLS   80
#define LDI    10240            // padded pred-stride of costT (per gt row)
#define IMG_W_F 1333.0f
#define IMG_H_F 800.0f
#define EPS_F  1e-12f
#define RADIUS 2.5f
#define MAX_ITERS 64

typedef __attribute__((ext_vector_type(2))) float v2f;
typedef __attribute__((ext_vector_type(8))) float v8f;

// CDNA5 async copy: global memory -> LDS (16B), tracked by ASYNCcnt.
// ldsaddr = low 32 bits of a generic pointer to __shared__ (wave-relative LDS offset).
__device__ __forceinline__ void async_copy_b128(void* lds_dst, const void* gsrc) {
  uint32_t la = (uint32_t)(size_t)lds_dst;
  asm volatile("global_load_async_to_lds_b128 %0, %1, off"
               :: "v"(la), "v"(gsrc) : "memory");
}
__device__ __forceinline__ void async_wait_all() {
  asm volatile("s_wait_asynccnt 0x0" ::: "memory");
}

// ---------------------------------------------------------------- init
__global__ __launch_bounds__(256) void init_kernel(unsigned int* __restrict__ mm32,
                                                   int* __restrict__ row_count,
                                                   int* __restrict__ col_count,
                                                   float* __restrict__ penalty,
                                                   int* __restrict__ flags) {
  int idx = blockIdx.x * 256 + threadIdx.x;
  const int MMW = (N_GT * N_PRED) / 4;   // 2,500,000 words
  if (idx < MMW) mm32[idx] = 0u;
  if (idx < N_PRED) { row_count[idx] = 0; penalty[idx] = 0.0f; }
  if (idx < N_GT) col_count[idx] = 0;
  if (idx < 2) flags[idx] = 0;
}

// ---------------------------------------------------------------- focal table D[i][c] = (pos-neg)*CLS_W
__global__ __launch_bounds__(256) void focal_kernel(const float* __restrict__ logits,
                                                    float* __restrict__ Dmat) {
  int t = blockIdx.x * 256 + threadIdx.x;
  if (t >= N_PRED * NCLS) return;
  float x = logits[t];
  float p = 1.0f / (1.0f + __expf(-x));
  float pos = -logf(p + EPS_F) * 0.25f * (1.0f - p) * (1.0f - p);
  float neg = -log1pf(-(p - EPS_F)) * 0.75f * p * p;
  Dmat[t] = (pos - neg) * 2.0f;   // CLS_W folded in
}

// ---------------------------------------------------------------- per-pred validity penalty (10000 if not near any gt)
__global__ __launch_bounds__(256) void valid_kernel(const float* __restrict__ pred_boxes,
                                                    const float* __restrict__ gt_boxes,
                                                    float* __restrict__ rowpen) {
  __shared__ float gbs[N_GT * 4];
  // async stage: 1000 gt boxes (16 KB) straight into LDS
  for (int t = threadIdx.x; t < N_GT; t += 256)
    async_copy_b128(&gbs[t * 4], gt_boxes + t * 4);
  async_wait_all();
  __syncthreads();
  int i = blockIdx.x * 256 + threadIdx.x;
  if (i >= N_PRED) return;
  float4 pbx = ((const float4*)pred_boxes)[i];
  float pcx = (pbx.x + pbx.z) * 0.5f, pcy = (pbx.y + pbx.w) * 0.5f;
  bool anyv = false;
  for (int g = 0; g < N_GT; g++) {
    float gx1 = gbs[g*4+0], gy1 = gbs[g*4+1], gx2 = gbs[g*4+2], gy2 = gbs[g*4+3];
    bool inb = (pcx > gx1) && (pcx < gx2) && (pcy > gy1) && (pcy < gy2);
    float gcx = (gx1 + gx2) * 0.5f, gcy = (gy1 + gy2) * 0.5f;
    float gw = gx2 - gx1, gh = gy2 - gy1;
    bool inc = (pcx > gcx - RADIUS*gw) && (pcx < gcx + RADIUS*gw) &&
               (pcy > gcy - RADIUS*gh) && (pcy < gcy + RADIUS*gh);
    if (inb || inc) { anyv = true; break; }
  }
  rowpen[i] = anyv ? 0.0f : 10000.0f;
}

// ---------------------------------------------------------------- fused WMMA cost kernel
// cls = D(16x80) @ onehot(labels)(80x16) via V_WMMA_F32_16X16X4_F32, K=80 -> 20 wmma ops.
// C/D fragment: lane holds gt column n=(lane&15), pred rows m = 8*(lane>>4)+r for VGPR r.
// Staging uses CDNA5 async global->LDS b128 copies; geometric terms fused in epilogue.
__global__ __launch_bounds__(256) void cost_wmma_kernel(const float* __restrict__ Dmat,
                                                        const float* __restrict__ pred_boxes,
                                                        const float* __restrict__ gt_boxes,
                                                        const int*   __restrict__ gt_labels,
                                                        const float* __restrict__ rowpen,
                                                        float* __restrict__ costT) {
  __shared__ float Asub[16 * NCLS];   // 16 pred rows x 80 classes (rows contiguous in Dmat)
  __shared__ float pb[16 * 4];        // 16 pred boxes
  __shared__ float gb[128 * 4];       // 128 gt boxes
  __shared__ int   glab[128];
  __shared__ float rp[16];

  const int tid = threadIdx.x;
  const int i0 = blockIdx.x * 16;     // pred tile base (625 blocks, exact)
  const int g0 = blockIdx.y * 128;    // gt tile base (8 blocks, guarded)

  // async stage Asub (1280 floats, linear copy), pb (64), rp (16), gb (128 boxes)
  for (int t = tid; t < 320; t += 256)
    async_copy_b128(&Asub[t * 4], Dmat + (size_t)i0 * NCLS + t * 4);
  if (tid < 16) async_copy_b128(&pb[tid * 4], pred_boxes + (size_t)(i0 + tid) * 4);
  if (tid < 4)  async_copy_b128(&rp[tid * 4], rowpen + i0 + tid * 4);
  if (tid < 128) {
    int g = g0 + tid;
    int gs = (g < N_GT) ? g : 0;          // clamp: padded lanes never store
    async_copy_b128(&gb[tid * 4], gt_boxes + (size_t)gs * 4);
    glab[tid] = (g < N_GT) ? gt_labels[g] : -1;
  }
  async_wait_all();
  __syncthreads();

  const int w   = tid >> 5;           // wave 0..7 -> 16-col subtile
  const int l   = tid & 31;
  const int n   = l & 15;
  const int hi  = l >> 4;             // 0/1
  const int koff = hi * 2;
  const int gloc = w * 16 + n;        // 0..127
  const int lab  = glab[gloc];

  v8f acc = {0.f, 0.f, 0.f, 0.f, 0.f, 0.f, 0.f, 0.f};
#pragma unroll
  for (int k = 0; k < NCLS; k += 4) {
    v2f a, b;
    a.x = Asub[n * NCLS + k + koff];
    a.y = Asub[n * NCLS + k + koff + 1];
    b.x = (lab == (k + koff))     ? 1.0f : 0.0f;
    b.y = (lab == (k + koff + 1)) ? 1.0f : 0.0f;
    acc = __builtin_amdgcn_wmma_f32_16x16x4_f32(false, a, false, b,
                                                (short)0, acc, false, false);
  }

  const int g = g0 + gloc;
  const float gx1 = gb[gloc*4+0], gy1 = gb[gloc*4+1];
  const float gx2 = gb[gloc*4+2], gy2 = gb[gloc*4+3];
  const float area_g = (gx2 - gx1) * (gy2 - gy1);
  const float gcx = (gx1 + gx2) * 0.5f, gcy = (gy1 + gy2) * 0.5f;
  const float gw = gx2 - gx1, gh = gy2 - gy1;
  const float invW = 5.0f / IMG_W_F, invH = 5.0f / IMG_H_F;  // L1_W folded

  float res[8];
#pragma unroll
  for (int r = 0; r < 8; r++) {
    int m = hi * 8 + r;
    float px1 = pb[m*4+0], py1 = pb[m*4+1], px2 = pb[m*4+2], py2 = pb[m*4+3];
    float l1 = (fabsf(px1 - gx1) + fabsf(px2 - gx2)) * invW +
               (fabsf(py1 - gy1) + fabsf(py2 - gy2)) * invH;
    float iw = fmaxf(fminf(px2, gx2) - fmaxf(px1, gx1), 0.0f);
    float ih = fmaxf(fminf(py2, gy2) - fmaxf(py1, gy1), 0.0f);
    float inter = iw * ih;
    float uni = (px2 - px1) * (py2 - py1) + area_g - inter;
    float iou = inter / fmaxf(uni, EPS_F);
    float ew = fmaxf(fmaxf(px2, gx2) - fminf(px1, gx1), 0.0f);
    float eh = fmaxf(fmaxf(py2, gy2) - fminf(py1, gy1), 0.0f);
    float enc = ew * eh;
    float giou = iou - (enc - uni) / fmaxf(enc, EPS_F);
    float pcx = (px1 + px2) * 0.5f, pcy = (py1 + py2) * 0.5f;
    bool inb = (pcx > gx1) && (pcx < gx2) && (pcy > gy1) && (pcy < gy2);
    bool inc = (pcx > gcx - RADIUS*gw) && (pcx < gcx + RADIUS*gw) &&
               (pcy > gcy - RADIUS*gh) && (pcy < gcy + RADIUS*gh);
    float cpen = (inb && inc) ? 0.0f : 100.0f;
    res[r] = acc[r] + l1 + (-giou * 2.0f) + cpen + rp[m];
  }
  if (g < N_GT) {
    float4* dst = (float4*)(costT + (size_t)g * LDI + i0 + hi * 8);
    dst[0] = make_float4(res[0], res[1], res[2], res[3]);
    dst[1] = make_float4(res[4], res[5], res[6], res[7]);
  }
}

// ---------------------------------------------------------------- per-GT top-5 (cost asc + iou desc) -> initial mm
__global__ __launch_bounds__(256) void topk_kernel(const float* __restrict__ costT,
                                                   const float* __restrict__ pred_boxes,
                                                   const float* __restrict__ gt_boxes,
                                                   unsigned char* __restrict__ mmT,
                                                   int* __restrict__ row_count,
                                                   int* __restrict__ col_count) {
  __shared__ float sc[256 * 5];
  __shared__ int   si[256 * 5];
  __shared__ float sv[256 * 5];
  const int g = blockIdx.x;
  const int tid = threadIdx.x;
  const float gx1 = gt_boxes[g*4+0], gy1 = gt_boxes[g*4+1];
  const float gx2 = gt_boxes[g*4+2], gy2 = gt_boxes[g*4+3];
  const float area_g = (gx2 - gx1) * (gy2 - gy1);
  float c5[5] = {FLT_MAX, FLT_MAX, FLT_MAX, FLT_MAX, FLT_MAX};
  int   i5[5] = {0x7fffffff, 0x7fffffff, 0x7fffffff, 0x7fffffff, 0x7fffffff};
  float v5[5] = {-1.f, -1.f, -1.f, -1.f, -1.f};
  const float* crow = costT + (size_t)g * LDI;
  const float4* pb4 = (const float4*)pred_boxes;
  for (int i = tid; i < N_PRED; i += 256) {
    __builtin_prefetch(crow + i + 2048, 0, 0);
    float cv = crow[i];
    if (cv < c5[4] || (cv == c5[4] && i < i5[4])) {
      c5[4] = cv; i5[4] = i;
#pragma unroll
      for (int j = 4; j > 0; j--) {
        bool sw = (c5[j] < c5[j-1]) || (c5[j] == c5[j-1] && i5[j] < i5[j-1]);
        if (sw) { float tc = c5[j]; c5[j] = c5[j-1]; c5[j-1] = tc;
                  int ti = i5[j]; i5[j] = i5[j-1]; i5[j-1] = ti; }
      }
    }
    float4 p = pb4[i];
    float iw = fmaxf(fminf(p.z, gx2) - fmaxf(p.x, gx1), 0.0f);
    float ih = fmaxf(fminf(p.w, gy2) - fmaxf(p.y, gy1), 0.0f);
    float inter = iw * ih;
    float uni = (p.z - p.x) * (p.w - p.y) + area_g - inter;
    float iou = inter / fmaxf(uni, EPS_F);
    if (iou > v5[4]) {
      v5[4] = iou;
#pragma unroll
      for (int j = 4; j > 0; j--)
        if (v5[j] > v5[j-1]) { float t = v5[j]; v5[j] = v5[j-1]; v5[j-1] = t; }
    }
  }
#pragma unroll
  for (int j = 0; j < 5; j++) { sc[tid*5+j] = c5[j]; si[tid*5+j] = i5[j]; sv[tid*5+j] = v5[j]; }
  for (int off = 128; off > 0; off >>= 1) {
    __syncthreads();
    if (tid < off) {
      float ac[5], bc[5], rc[5]; int ai[5], bi[5], ri[5];
#pragma unroll
      for (int j = 0; j < 5; j++) { ac[j] = sc[tid*5+j]; ai[j] = si[tid*5+j];
                                    bc[j] = sc[(tid+off)*5+j]; bi[j] = si[(tid+off)*5+j]; }
      int x = 0, y = 0;
#pragma unroll
      for (int j = 0; j < 5; j++) {
        bool ta = (y >= 5) || (x < 5 && ((ac[x] < bc[y]) || (ac[x] == bc[y] && ai[x] <= bi[y])));
        if (ta) { rc[j] = ac[x]; ri[j] = ai[x]; x++; } else { rc[j] = bc[y]; ri[j] = bi[y]; y++; }
      }
#pragma unroll
      for (int j = 0; j < 5; j++) { sc[tid*5+j] = rc[j]; si[tid*5+j] = ri[j]; }
      float av[5], bv[5], rv[5];
#pragma unroll
      for (int j = 0; j < 5; j++) { av[j] = sv[tid*5+j]; bv[j] = sv[(tid+off)*5+j]; }
      x = 0; y = 0;
#pragma unroll
      for (int j = 0; j < 5; j++) {
        bool ta = (y >= 5) || (x < 5 && (av[x] >= bv[y]));
        if (ta) { rv[j] = av[x]; x++; } else { rv[j] = bv[y]; y++; }
      }
#pragma unroll
      for (int j = 0; j < 5; j++) sv[tid*5+j] = rv[j];
    }
  }
  __syncthreads();
  if (tid == 0) {
    float s = sv[0] + sv[1] + sv[2] + sv[3] + sv[4];
    int dk = (int)s; if (dk < 1) dk = 1; if (dk > 5) dk = 5;
    for (int j = 0; j < dk; j++) {
      int i = si[j];
      mmT[(size_t)g * N_PRED + i] = 1;
      atomicAdd(&row_count[i], 1);
    }
    col_count[g] = dk;
  }
}

// ---------------------------------------------------------------- resolve multi-matched preds (prior_mask) once
__global__ __launch_bounds__(256) void prior_kernel(const float* __restrict__ costT,
                                                    unsigned char* __restrict__ mmT,
                                                    int* __restrict__ row_count,
                                                    int* __restrict__ col_count,
                                                    unsigned char* __restrict__ prior,
                                                    int* __restrict__ cminA) {
  int i = blockIdx.x * 256 + threadIdx.x;
  if (i >= N_PRED) return;
  int rc = row_count[i];
  prior[i] = (rc > 1) ? 1 : 0;
  if (rc <= 1) return;
  float best = FLT_MAX; int bg = 0;
  for (int g = 0; g < N_GT; g++) {
    float c = costT[(size_t)g * LDI + i];
    if (c < best) { best = c; bg = g; }
  }
  cminA[i] = bg;
  for (int g = 0; g < N_GT; g++) {
    unsigned char v = mmT[(size_t)g * N_PRED + i];
    if (g == bg) {
      if (!v) { mmT[(size_t)g * N_PRED + i] = 1; atomicAdd(&col_count[g], 1); }
    } else if (v) {
      mmT[(size_t)g * N_PRED + i] = 0; atomicSub(&col_count[g], 1);
    }
  }
  row_count[i] = 1;
}

// ---------------------------------------------------------------- matching loop kernels (flag-gated)
__global__ __launch_bounds__(1024) void check_kernel(const int* __restrict__ col_count,
                                                     int* __restrict__ flags) {
  if (flags[0]) return;
  __shared__ int anyu;
  if (threadIdx.x == 0) anyu = 0;
  __syncthreads();
  if (threadIdx.x < N_GT && col_count[threadIdx.x] == 0) atomicOr(&anyu, 1);
  __syncthreads();
  if (threadIdx.x == 0 && !anyu) flags[0] = 1;   // converged
}

__global__ __launch_bounds__(256) void penalty_kernel(const int* __restrict__ row_count,
                                                      float* __restrict__ penalty,
                                                      int* __restrict__ flags) {
  if (flags[0]) return;
  int i = blockIdx.x * 256 + threadIdx.x;
  if (i == 0) flags[1] = 0;     // reset anyrow for this iteration
  if (i < N_PRED && row_count[i] > 0) penalty[i] += 100000.0f;
}

__global__ __launch_bounds__(256) void assign_kernel(const float* __restrict__ costT,
                                                     const float* __restrict__ penalty,
                                                     unsigned char* __restrict__ mmT,
                                                     int* __restrict__ row_count,
                                                     int* __restrict__ col_count,
                                                     const int* __restrict__ flags) {
  if (flags[0]) return;
  const int g = blockIdx.x;
  if (col_count[g] != 0) return;      // only unmatched columns
  __shared__ float sval[256];
  __shared__ int   sidx[256];
  const int tid = threadIdx.x;
  const float* crow = costT + (size_t)g * LDI;
  float best = FLT_MAX; int bi = 0x7fffffff;
  for (int i = tid; i < N_PRED; i += 256) {
    __builtin_prefetch(crow + i + 2048, 0, 0);
    float c = crow[i] + penalty[i];
    if (c < best) { best = c; bi = i; }
  }
  sval[tid] = best; sidx[tid] = bi;
  __syncthreads();
  for (int off = 128; off > 0; off >>= 1) {
    if (tid < off) {
      float v2 = sval[tid + off]; int i2 = sidx[tid + off];
      if (v2 < sval[tid] || (v2 == sval[tid] && i2 < sidx[tid])) { sval[tid] = v2; sidx[tid] = i2; }
    }
    __syncthreads();
  }
  if (tid == 0) {
    int pos = sidx[0];
    mmT[(size_t)g * N_PRED + pos] = 1;
    atomicAdd(&row_count[pos], 1);
    col_count[g] = 1;
  }
}

__global__ __launch_bounds__(256) void anyrow_kernel(const int* __restrict__ row_count,
                                                     int* __restrict__ flags) {
  if (flags[0]) return;
  int i = blockIdx.x * 256 + threadIdx.x;
  if (i < N_PRED && row_count[i] > 1) atomicOr(&flags[1], 1);
}

__global__ __launch_bounds__(256) void fix_kernel(unsigned char* __restrict__ mmT,
                                                  int* __restrict__ row_count,
                                                  int* __restrict__ col_count,
                                                  const unsigned char* __restrict__ prior,
                                                  const int* __restrict__ cminA,
                                                  const int* __restrict__ flags) {
  if (flags[0] || !flags[1]) return;
  int i = blockIdx.x * 256 + threadIdx.x;
  if (i >= N_PRED || !prior[i]) return;
  int bg = cminA[i];
  for (int g = 0; g < N_GT; g++) {
    unsigned char v = mmT[(size_t)g * N_PRED + i];
    if (g == bg) {
      if (!v) { mmT[(size_t)g * N_PRED + i] = 1; atomicAdd(&col_count[g], 1); }
    } else if (v) {
      mmT[(size_t)g * N_PRED + i] = 0; atomicSub(&col_count[g], 1);
    }
  }
  row_count[i] = 1;
}

// ---------------------------------------------------------------- finalize: fg_mask + matched_gt
__global__ __launch_bounds__(256) void final_kernel(const unsigned char* __restrict__ mmT,
                                                    const int* __restrict__ row_count,
                                                    float* __restrict__ out) {
  int i = blockIdx.x * 256 + threadIdx.x;
  if (i >= N_PRED) return;
  int fg = row_count[i] > 0;
  int mg = 0;
  if (fg) {
    for (int g = 0; g < N_GT; g++)
      if (mmT[(size_t)g * N_PRED + i]) { mg = g; break; }
  }
  out[i] = fg ? 1.0f : 0.0f;
  out[N_PRED + i] = (float)mg;
}

// ---------------------------------------------------------------- host launcher
extern "C" void kernel_launch(void* const* d_in, const int* in_sizes, int n_in,
                              void* d_out, int out_size, void* d_ws, size_t ws_size,
                              hipStream_t stream) {
  const float* logits  = (const float*)d_in[0];
  const float* pboxes  = (const float*)d_in[1];
  const float* gboxes  = (const float*)d_in[2];
  const int*   glabels = (const int*)d_in[3];
  float* out = (float*)d_out;

  char* base = (char*)d_ws;
  size_t off = 0;
  auto take = [&](size_t bytes) -> char* {
    char* p = base + off;
    off = (off + bytes + 255) & ~(size_t)255;
    return p;
  };
  float*         costT   = (float*)take((size_t)N_GT * LDI * sizeof(float));   // 40.96 MB
  float*         Dmat    = (float*)take((size_t)N_PRED * NCLS * sizeof(float));// 3.2 MB
  unsigned char* mmT     = (unsigned char*)take((size_t)N_GT * N_PRED);        // 10 MB
  float*         rowpen  = (float*)take(N_PRED * sizeof(float));
  float*         penalty = (float*)take(N_PRED * sizeof(float));
  int*           row_cnt = (int*)take(N_PRED * sizeof(int));
  int*           col_cnt = (int*)take(N_GT * sizeof(int));
  int*           cminA   = (int*)take(N_PRED * sizeof(int));
  unsigned char* prior   = (unsigned char*)take(N_PRED);
  int*           flags   = (int*)take(2 * sizeof(int));

  const int PB = (N_PRED + 255) / 256;                 // 40
  init_kernel<<<((N_GT * N_PRED / 4) + 255) / 256, 256, 0, stream>>>(
      (unsigned int*)mmT, row_cnt, col_cnt, penalty, flags);
  focal_kernel<<<(N_PRED * NCLS + 255) / 256, 256, 0, stream>>>(logits, Dmat);
  valid_kernel<<<PB, 256, 0, stream>>>(pboxes, gboxes, rowpen);
  cost_wmma_kernel<<<dim3(N_PRED / 16, (N_GT + 127) / 128), 256, 0, stream>>>(
      Dmat, pboxes, gboxes, glabels, rowpen, costT);
  topk_kernel<<<N_GT, 256, 0, stream>>>(costT, pboxes, gboxes, mmT, row_cnt, col_cnt);
  prior_kernel<<<PB, 256, 0, stream>>>(costT, mmT, row_cnt, col_cnt, prior, cminA);
  for (int it = 0; it < MAX_ITERS; ++it) {
    check_kernel<<<1, 1024, 0, stream>>>(col_cnt, flags);
    penalty_kernel<<<PB, 256, 0, stream>>>(row_cnt, penalty, flags);
    assign_kernel<<<N_GT, 256, 0, stream>>>(costT, penalty, mmT, row_cnt, col_cnt, flags);
    anyrow_kernel<<<PB, 256, 0, stream>>>(row_cnt, flags);
    fix_kernel<<<PB, 256, 0, stream>>>(mmT, row_cnt, col_cnt, prior, cminA, flags);
  }
  final_kernel<<<PB, 256, 0, stream>>>(mmT, row_cnt, out);

  (void)in_sizes; (void)n_in; (void)out_size; (void)ws_size;
}